// SL2Net_43293270344155
// MI455X (gfx1250) — compile-verified
//
#include <hip/hip_runtime.h>
#include <cstdint>
#include <cstddef>

typedef float v2f __attribute__((ext_vector_type(2)));
typedef float v8f __attribute__((ext_vector_type(8)));

#define NTH 256
#define NBATCH 16

namespace sl2 {

constexpr int BTOT = 16384;
constexpr int OUTS = 45;                       // sum of (d+1), d=0..8
constexpr int DOFF[5] = {0, 1, 3, 6, 10};      // offsets of xn[d] inside the 15-float register pack
constexpr int CNT[9]  = {6, 5, 8, 6, 7, 3, 3, 1, 1};   // feature groups per degree
constexpr int SOFF[9] = {0, 1, 3, 6, 10, 15, 21, 28, 36};
constexpr int NPG[9]  = {5, 4, 7, 5, 6, 3, 3, 1, 1};   // pair-groups per degree
constexpr int HASSELF[9] = {1, 1, 1, 1, 1, 0, 0, 0, 0};
// pair groups per degree, in contrib append order: {i, j, row offset inside irr}
constexpr int PG[9][7][3] = {
  {{0,0,0},{1,1,3},{2,2,8},{3,3,15},{4,4,24},{0,0,0},{0,0,0}},
  {{0,1,0},{1,2,4},{2,3,10},{3,4,18},{0,0,0},{0,0,0},{0,0,0}},
  {{0,2,0},{1,1,0},{1,3,5},{2,2,5},{2,4,12},{3,3,12},{4,4,21}},
  {{0,3,0},{1,2,0},{1,4,6},{2,3,6},{3,4,14},{0,0,0},{0,0,0}},
  {{0,4,0},{1,3,0},{2,2,0},{2,4,7},{3,3,7},{4,4,16},{0,0,0}},
  {{1,4,0},{2,3,0},{3,4,8},{0,0,0},{0,0,0},{0,0,0},{0,0,0}},
  {{2,4,0},{3,3,0},{4,4,9},{0,0,0},{0,0,0},{0,0,0},{0,0,0}},
  {{3,4,0},{0,0,0},{0,0,0},{0,0,0},{0,0,0},{0,0,0},{0,0,0}},
  {{4,4,0},{0,0,0},{0,0,0},{0,0,0},{0,0,0},{0,0,0},{0,0,0}}
};
// flat offsets of each pair's T matrix inside the 2002-float T pool
constexpr int TOFF[5][5] = {
  {0,    1,    5,   14,   30},
  {0,   55,   71,  107,  171},
  {0,    0,  271,  352,  496},
  {0,    0,    0,  721,  977},
  {0,    0,    0,    0, 1377}
};
constexpr int TTOT = 2002;
constexpr int FEATS_LDS = 35840;   // max rows*stride over degrees (deg4: 448*80)

// k-major weight pool layout inside ws (element offsets from wt base)
constexpr int KDEG[9]  = {384, 320, 512, 384, 448, 192, 192, 64, 64};
constexpr int WTOFF[9] = {0, 24576, 45056, 77824, 102400, 131072, 143360, 155648, 159744};
constexpr int WT_DEG_TOT = 163840;
constexpr int WT_M1 = WT_DEG_TOT;            // w1t: [384][128]
constexpr int WT_M2 = WT_M1 + 49152;         // w2t: [128][128]
constexpr int WT_M3 = WT_M2 + 16384;         // w3t: [128][384]
constexpr int WT_TOT = WT_M3 + 49152;        // 278528 floats

__device__ __forceinline__ v8f wmma_f32(v2f a, v2f b, v8f c) {
  // D(16x16,f32) = A(16x4,f32) x B(4x16,f32) + C
  return __builtin_amdgcn_wmma_f32_16x16x4_f32(false, a, false, b, (short)0, c,
                                               false, false);
}

// prefetch a k-major weight tile into cache (one byte per 256B region)
__device__ __forceinline__ void prefetch_w(const float* p, int nfloat, int tid) {
  for (int off = tid * 64; off < nfloat; off += NTH * 64)
    __builtin_prefetch(p + off, 0, 3);
}

// ---------------- producer: build feats[k][col] for one degree in LDS -------
template<int D>
__device__ __forceinline__ void produce(const float (&xr)[4][15], float* feats,
                                        const float* Tlds, int tid) {
  constexpr int NC = 16 * (D + 1);
  constexpr int ST = (NC % 64 == 0) ? NC + 8 : NC;   // LDS row stride (bank-pad)
#pragma unroll
  for (int rsl = 0; rsl < 4; ++rsl) {
    const int bc = tid + 256 * rsl;          // 0..1023 = (b_local, c)
    const int bl = bc >> 6, c = bc & 63;
    const float* x = xr[rsl];
    const int colb = bl * (D + 1);
    if (HASSELF[D]) {
#pragma unroll
      for (int s = 0; s <= D; ++s)
        feats[c * ST + colb + s] = x[DOFF[D] + s];
    }
#pragma unroll
    for (int pg = 0; pg < NPG[D]; ++pg) {
      const int i = PG[D][pg][0], j = PG[D][pg][1], r = PG[D][pg][2];
      const int m = i + 1, n = j + 1, mn = m * n;
      const int g = (HASSELF[D] ? 1 : 0) + pg;
      float op[25];
#pragma unroll
      for (int p = 0; p < m; ++p)
#pragma unroll
        for (int q = 0; q < n; ++q)
          op[p * n + q] = x[DOFF[i] + p] * x[DOFF[j] + q];
      const float* Tp = Tlds + TOFF[i][j];
#pragma unroll
      for (int s = 0; s <= D; ++s) {
        const float* Trow = Tp + (r + s) * mn;
        float a = 0.f;
#pragma unroll
        for (int e = 0; e < mn; ++e) a = fmaf(Trow[e], op[e], a);
        feats[(g * 64 + c) * ST + colb + s] = a;
      }
    }
  }
}

// ------- consumer: out[b,o,s] = Wt_d[k,o] * feats[k,(b,s)], Wt k-major ------
template<int D>
__device__ __forceinline__ void consume(const float* feats, const float* __restrict__ Wt,
                                        float* __restrict__ out, int b0, int tid) {
  constexpr int K = 64 * CNT[D];
  constexpr int NCT = D + 1;                 // column tiles of 16
  constexpr int NC = 16 * NCT;
  constexpr int ST = (NC % 64 == 0) ? NC + 8 : NC;
  constexpr int MAXC = (NCT + 1) / 2;        // <=5 column tiles per wave
  const int lane = tid & 31, wave = tid >> 5;
  const int m0 = (wave & 3) * 16;            // output-channel tile
  const int ch = wave >> 2;                  // column half (0/1)
  const int r16 = lane & 15, kh = lane >> 4;
  v8f acc[MAXC];
#pragma unroll
  for (int ci = 0; ci < MAXC; ++ci)
#pragma unroll
    for (int v = 0; v < 8; ++v) acc[ci][v] = 0.f;
  const float* wp = Wt + m0 + r16;           // coalesced: lanes 0-15 -> 64B line
  for (int k0 = 0; k0 < K; k0 += 32) {
    // preload 8 A fragments (coalesced b32 loads) so waits amortize over 8 WMMAs
    v2f areg[8];
#pragma unroll
    for (int u = 0; u < 8; ++u) {
      const int kk = k0 + 4 * u + 2 * kh;
      areg[u].x = wp[(size_t)kk * 64];
      areg[u].y = wp[(size_t)(kk + 1) * 64];
    }
#pragma unroll
    for (int u = 0; u < 8; ++u) {
      const int kk = k0 + 4 * u + 2 * kh;
#pragma unroll
      for (int ci = 0; ci < MAXC; ++ci) {
        const int ct = ch + 2 * ci;
        if (ct < NCT) {
          const int col = ct * 16 + r16;
          v2f bf;                            // B: lane col, elem j -> K=kk+j
          bf.x = feats[kk * ST + col];
          bf.y = feats[(kk + 1) * ST + col];
          acc[ci] = wmma_f32(areg[u], bf, acc[ci]);
        }
      }
    }
  }
#pragma unroll
  for (int ci = 0; ci < MAXC; ++ci) {
    const int ct = ch + 2 * ci;
    if (ct < NCT) {
      const int col = ct * 16 + r16;
      const int bl = col / (D + 1);
      const int s = col - bl * (D + 1);
#pragma unroll
      for (int v = 0; v < 8; ++v) {
        const int o = m0 + v + 8 * kh;       // D frag: M = v + 8*lanehalf
        out[((size_t)(b0 + bl) * 64 + o) * OUTS + SOFF[D] + s] = acc[ci][v];
      }
    }
  }
}

// ------- MLP GEMM (LDS->LDS), k-major weights, bias + optional leaky-relu ---
template<int M, int K, bool RELU>
__device__ __forceinline__ void mlp_gemm(const float* __restrict__ Wt,
                                         const float* __restrict__ bias,
                                         const float* fin, float* fout, int tid) {
  const int lane = tid & 31, wave = tid >> 5;
  const int r16 = lane & 15, kh = lane >> 4;
  constexpr int MT = M / 16;
  for (int mt = wave; mt < MT; mt += 8) {
    const int m0 = mt * 16;
    v8f acc;
#pragma unroll
    for (int v = 0; v < 8; ++v) acc[v] = bias[m0 + v + 8 * kh];
    const float* wp = Wt + m0 + r16;
    for (int k0 = 0; k0 < K; k0 += 32) {
      v2f areg[8];
#pragma unroll
      for (int u = 0; u < 8; ++u) {
        const int kk = k0 + 4 * u + 2 * kh;
        areg[u].x = wp[(size_t)kk * M];
        areg[u].y = wp[(size_t)(kk + 1) * M];
      }
#pragma unroll
      for (int u = 0; u < 8; ++u) {
        const int kk = k0 + 4 * u + 2 * kh;
        v2f bf;
        bf.x = fin[kk * 16 + r16];
        bf.y = fin[(kk + 1) * 16 + r16];
        acc = wmma_f32(areg[u], bf, acc);
      }
    }
#pragma unroll
    for (int v = 0; v < 8; ++v) {
      float val = acc[v];
      if (RELU) val = (val > 0.f) ? val : 0.01f * val;
      fout[(m0 + v + 8 * kh) * 16 + r16] = val;
    }
  }
}

} // namespace sl2

// ======================= kernel 0: generate T matrices ======================
// Deterministic stand-in for np.random.default_rng(1000*m+n).standard_normal
// (splitmix64 + Box-Muller, scaled by 1/sqrt(m*n)).  On hardware these 8KB
// would be precomputed host-side with numpy and uploaded.
__global__ void __launch_bounds__(NTH) sl2_tgen(float* __restrict__ Tg) {
  const int PI[15] = {0,0,0,0,0,1,1,1,1,2,2,2,3,3,4};
  const int PJ[15] = {0,1,2,3,4,1,2,3,4,2,3,4,3,4,4};
  const int POFF[16] = {0,1,5,14,30,55,71,107,171,271,352,496,721,977,1377,2002};
  for (int e = threadIdx.x + blockIdx.x * blockDim.x; e < 2002;
       e += blockDim.x * gridDim.x) {
    int pi = 0;
    while (e >= POFF[pi + 1]) ++pi;
    const int le = e - POFF[pi];
    const int m = PI[pi] + 1, n = PJ[pi] + 1;
    unsigned long long z = (unsigned long long)(1000 * m + n + 1) * 0x9E3779B97F4A7C15ull +
                           (unsigned long long)(le + 1) * 0xBF58476D1CE4E5B9ull;
    z ^= z >> 30; z *= 0xBF58476D1CE4E5B9ull;
    z ^= z >> 27; z *= 0x94D049BB133111EBull;
    z ^= z >> 31;
    const unsigned int u1 = (unsigned int)(z >> 32), u2 = (unsigned int)z;
    const float f1 = ((float)u1 + 1.0f) * (1.0f / 4294967296.0f);
    const float f2 = (float)u2 * (6.2831853071795864769f / 4294967296.0f);
    const float val = sqrtf(-2.0f * logf(f1)) * cosf(f2);
    Tg[e] = val * rsqrtf((float)(m * n));
  }
}

// ============ kernel 1: per-(degree,channel) variance -> scale ==============
__global__ void __launch_bounds__(NTH) sl2_var(
    const float* __restrict__ x0, const float* __restrict__ x1,
    const float* __restrict__ x2, const float* __restrict__ x3,
    const float* __restrict__ x4, const float* __restrict__ bn0,
    const float* __restrict__ bn1, const float* __restrict__ bn2,
    const float* __restrict__ bn3, const float* __restrict__ bn4,
    float* __restrict__ scales) {
  const int d = blockIdx.x >> 6;
  const int c = blockIdx.x & 63;
  const float* xs[5] = {x0, x1, x2, x3, x4};
  const float* bns[5] = {bn0, bn1, bn2, bn3, bn4};
  const float* x = xs[d];
  const int P = d + 1;
  float sum = 0.f, sq = 0.f;
  for (int b = threadIdx.x; b < sl2::BTOT; b += NTH) {
    const float* px = x + ((size_t)b * 64 + c) * P;
    for (int p = 0; p < P; ++p) {
      const float v = px[p];
      sum += v;
      sq = fmaf(v, v, sq);
    }
  }
  __shared__ float s1[NTH], s2[NTH];
  s1[threadIdx.x] = sum;
  s2[threadIdx.x] = sq;
  __syncthreads();
  for (int o = NTH / 2; o > 0; o >>= 1) {
    if (threadIdx.x < o) {
      s1[threadIdx.x] += s1[threadIdx.x + o];
      s2[threadIdx.x] += s2[threadIdx.x + o];
    }
    __syncthreads();
  }
  if (threadIdx.x == 0) {
    const float N = (float)sl2::BTOT * (float)P;
    const float mean = s1[0] / N;
    const float var = s2[0] / N - mean * mean;
    scales[d * 64 + c] = bns[d][c] * rsqrtf(var + 1e-5f);
  }
}

// ======= kernel 2: transpose one (M,K) row-major weight to k-major [K][M] ===
__global__ void __launch_bounds__(NTH) sl2_wt(const float* __restrict__ src,
                                              float* __restrict__ dst,
                                              int M, int K) {
  const int idx = blockIdx.x * blockDim.x + threadIdx.x;  // coalesced read
  if (idx < M * K) {
    const int m = idx / K, k = idx - m * K;
    dst[(size_t)k * M + m] = src[idx];
  }
}

// ======== kernel 3: fused normalize + tensor-product + MLP + WMMA GEMMs =====
__global__ void __launch_bounds__(NTH) sl2_main(
    const float* __restrict__ x0, const float* __restrict__ x1,
    const float* __restrict__ x2, const float* __restrict__ x3,
    const float* __restrict__ x4, const float* __restrict__ Tg,
    const float* __restrict__ scales, const float* __restrict__ wt,
    const float* __restrict__ mb1, const float* __restrict__ mb2,
    const float* __restrict__ mb3, float* __restrict__ out) {
  __shared__ float Tlds[sl2::TTOT];
  __shared__ float sc[320];
  __shared__ float feats[sl2::FEATS_LDS];
  const int tid = threadIdx.x;
  for (int i = tid; i < sl2::TTOT; i += NTH) Tlds[i] = Tg[i];
  for (int i = tid; i < 320; i += NTH) sc[i] = scales[i];
  __syncthreads();

  const int b0 = blockIdx.x * NBATCH;
  const float* xs[5] = {x0, x1, x2, x3, x4};
  float xr[4][15];  // 4 (b,c) slots * 15 normalized components, in registers
#pragma unroll
  for (int rsl = 0; rsl < 4; ++rsl) {
    const int bc = tid + 256 * rsl;
    const int bl = bc >> 6, c = bc & 63;
    const size_t base = (size_t)(b0 + bl) * 64 + c;
#pragma unroll
    for (int d = 0; d < 5; ++d) {
      const float s = sc[d * 64 + c];
#pragma unroll
      for (int p = 0; p <= d; ++p)
        xr[rsl][sl2::DOFF[d] + p] = xs[d][base * (d + 1) + p] * s;
    }
  }

  // ---- degree 0 (with MLP) ----
  sl2::prefetch_w(wt + sl2::WT_M1, sl2::WT_TOT - sl2::WT_M1, tid);  // MLP weights
  sl2::prefetch_w(wt + sl2::WTOFF[0], 64 * sl2::KDEG[0], tid);
  sl2::produce<0>(xr, feats, Tlds, tid);          // feats0: [384][16] at offset 0
  __syncthreads();
  sl2::mlp_gemm<128, 384, true>(wt + sl2::WT_M1, mb1, feats, feats + 8192, tid);
  __syncthreads();
  sl2::mlp_gemm<128, 128, true>(wt + sl2::WT_M2, mb2, feats + 8192, feats + 10240, tid);
  __syncthreads();
  sl2::mlp_gemm<384, 128, false>(wt + sl2::WT_M3, mb3, feats + 10240, feats, tid);
  __syncthreads();
  sl2::consume<0>(feats, wt + sl2::WTOFF[0], out, b0, tid);
  __syncthreads();

  // ---- degrees 1..8: prefetch Wt during producer VALU phase ----
  sl2::prefetch_w(wt + sl2::WTOFF[1], 64 * sl2::KDEG[1], tid);
  sl2::produce<1>(xr, feats, Tlds, tid); __syncthreads();
  sl2::consume<1>(feats, wt + sl2::WTOFF[1], out, b0, tid); __syncthreads();
  sl2::prefetch_w(wt + sl2::WTOFF[2], 64 * sl2::KDEG[2], tid);
  sl2::produce<2>(xr, feats, Tlds, tid); __syncthreads();
  sl2::consume<2>(feats, wt + sl2::WTOFF[2], out, b0, tid); __syncthreads();
  sl2::prefetch_w(wt + sl2::WTOFF[3], 64 * sl2::KDEG[3], tid);
  sl2::produce<3>(xr, feats, Tlds, tid); __syncthreads();
  sl2::consume<3>(feats, wt + sl2::WTOFF[3], out, b0, tid); __syncthreads();
  sl2::prefetch_w(wt + sl2::WTOFF[4], 64 * sl2::KDEG[4], tid);
  sl2::produce<4>(xr, feats, Tlds, tid); __syncthreads();
  sl2::consume<4>(feats, wt + sl2::WTOFF[4], out, b0, tid); __syncthreads();
  sl2::prefetch_w(wt + sl2::WTOFF[5], 64 * sl2::KDEG[5], tid);
  sl2::produce<5>(xr, feats, Tlds, tid); __syncthreads();
  sl2::consume<5>(feats, wt + sl2::WTOFF[5], out, b0, tid); __syncthreads();
  sl2::prefetch_w(wt + sl2::WTOFF[6], 64 * sl2::KDEG[6], tid);
  sl2::produce<6>(xr, feats, Tlds, tid); __syncthreads();
  sl2::consume<6>(feats, wt + sl2::WTOFF[6], out, b0, tid); __syncthreads();
  sl2::prefetch_w(wt + sl2::WTOFF[7], 64 * sl2::KDEG[7], tid);
  sl2::produce<7>(xr, feats, Tlds, tid); __syncthreads();
  sl2::consume<7>(feats, wt + sl2::WTOFF[7], out, b0, tid); __syncthreads();
  sl2::prefetch_w(wt + sl2::WTOFF[8], 64 * sl2::KDEG[8], tid);
  sl2::produce<8>(xr, feats, Tlds, tid); __syncthreads();
  sl2::consume<8>(feats, wt + sl2::WTOFF[8], out, b0, tid);
}

// ============================== launcher ====================================
extern "C" void kernel_launch(void* const* d_in, const int* in_sizes, int n_in,
                              void* d_out, int out_size, void* d_ws, size_t ws_size,
                              hipStream_t stream) {
  (void)in_sizes; (void)n_in; (void)out_size; (void)ws_size;
  const float* x0  = (const float*)d_in[0];
  const float* bn0 = (const float*)d_in[1];
  const float* x1  = (const float*)d_in[2];
  const float* bn1 = (const float*)d_in[3];
  const float* x2  = (const float*)d_in[4];
  const float* bn2 = (const float*)d_in[5];
  const float* x3  = (const float*)d_in[6];
  const float* bn3 = (const float*)d_in[7];
  const float* x4  = (const float*)d_in[8];
  const float* bn4 = (const float*)d_in[9];
  const float* mw1 = (const float*)d_in[10];
  const float* mb1 = (const float*)d_in[11];
  const float* mw2 = (const float*)d_in[12];
  const float* mb2 = (const float*)d_in[13];
  const float* mw3 = (const float*)d_in[14];
  const float* mb3 = (const float*)d_in[15];
  const float* wd[9];
  for (int d = 0; d < 9; ++d) wd[d] = (const float*)d_in[16 + d];
  float* out = (float*)d_out;

  // ws layout (floats): [0,2002) T | [2048,2368) scales | [2432, +278528) Wt pool
  float* Tg = (float*)d_ws;
  float* scales = Tg + 2048;
  float* wt = Tg + 2432;

  sl2_tgen<<<8, NTH, 0, stream>>>(Tg);
  sl2_var<<<320, NTH, 0, stream>>>(x0, x1, x2, x3, x4, bn0, bn1, bn2, bn3, bn4,
                                   scales);
  // transpose the 9 degree weights (64 x K) and 3 MLP weights to k-major
  for (int d = 0; d < 9; ++d) {
    const int K = sl2::KDEG[d];
    sl2_wt<<<(64 * K + NTH - 1) / NTH, NTH, 0, stream>>>(wd[d], wt + sl2::WTOFF[d],
                                                         64, K);
  }
  sl2_wt<<<(128 * 384 + NTH - 1) / NTH, NTH, 0, stream>>>(mw1, wt + sl2::WT_M1,
                                                          128, 384);
  sl2_wt<<<(128 * 128 + NTH - 1) / NTH, NTH, 0, stream>>>(mw2, wt + sl2::WT_M2,
                                                          128, 128);
  sl2_wt<<<(384 * 128 + NTH - 1) / NTH, NTH, 0, stream>>>(mw3, wt + sl2::WT_M3,
                                                          384, 128);
  sl2_main<<<sl2::BTOT / NBATCH, NTH, 0, stream>>>(
      x0, x1, x2, x3, x4, Tg, scales, wt, mb1, mb2, mb3, out);
}